// LightGCN_64476049047989
// MI455X (gfx1250) — compile-verified
//
#include <hip/hip_runtime.h>
#include <math.h>

#define NUSR 200000
#define NITM 100000
#define NTOT 300000
#define DIM  64
#define NE   1000000
#define NE2  2000000
#define HML  32
#define BBAT 4096

typedef float v2f __attribute__((ext_vector_type(2)));
typedef float v8f __attribute__((ext_vector_type(8)));

// ---------------- generic helpers ----------------
__global__ void zero_kernel(float* p, long n) {
    long i = (long)blockIdx.x * blockDim.x + threadIdx.x;
    long stride = (long)gridDim.x * blockDim.x;
    for (; i < n; i += stride) p[i] = 0.0f;
}

__global__ void copy_kernel(const float* __restrict__ src, float* __restrict__ dst, long n4) {
    // n4 = number of float4 elements
    long i = (long)blockIdx.x * blockDim.x + threadIdx.x;
    long stride = (long)gridDim.x * blockDim.x;
    const float4* s = (const float4*)src;
    float4* d = (float4*)dst;
    for (; i < n4; i += stride) d[i] = s[i];
}

// ---------------- edge weight MLP: sigmoid(W2 relu(W1 f + b1) + b2) ----------------
__global__ void edge_mlp_kernel(const float* __restrict__ ef,
                                const float* __restrict__ W1, const float* __restrict__ b1,
                                const float* __restrict__ W2, const float* __restrict__ b2,
                                float* __restrict__ w) {
    __shared__ float sW1[HML * 8];
    __shared__ float sb1[HML];
    __shared__ float sW2[HML];
    __shared__ float sb2;
    int tid = threadIdx.x;
    if (tid < HML * 8) sW1[tid] = W1[tid];
    if (tid < HML) { sb1[tid] = b1[tid]; sW2[tid] = W2[tid]; }
    if (tid == 0) sb2 = b2[0];
    __syncthreads();
    int e = blockIdx.x * blockDim.x + tid;
    if (e >= NE2) return;
    const float4* efp = (const float4*)(ef + (long)e * 8);
    float4 f0 = efp[0], f1 = efp[1];
    float f[8] = {f0.x, f0.y, f0.z, f0.w, f1.x, f1.y, f1.z, f1.w};
    float s = sb2;
    for (int j = 0; j < HML; ++j) {
        float h = sb1[j];
#pragma unroll
        for (int k = 0; k < 8; ++k) h = fmaf(sW1[j * 8 + k], f[k], h);
        h = fmaxf(h, 0.0f);
        s = fmaf(sW2[j], h, s);
    }
    w[e] = 1.0f / (1.0f + __expf(-s));
}

// ---------------- item fuse + adapter GEMM (V_WMMA_F32_16X16X4_F32) + L2 norm ----------------
__global__ void __launch_bounds__(32)
fuse_adapter_kernel(const float* __restrict__ item_audio,
                    const float* __restrict__ artist_emb,
                    const float* __restrict__ album_emb,
                    const int* __restrict__ artist_ids,
                    const int* __restrict__ album_ids,
                    const float* __restrict__ Wt,    // adapter_W [out=64, in=64] row-major
                    const float* __restrict__ bias,
                    float* __restrict__ x)           // writes rows [NUSR, NUSR+NITM)
{
    __shared__ float sA[16 * 64];
    __shared__ int   sAid[16], sBid[16];
    __shared__ float sInv[16];
    int lane = threadIdx.x;        // wave32: one wave per block
    int i0 = blockIdx.x * 16;      // 16 items per wave; NITM % 16 == 0
    if (lane < 16) {
        sAid[lane] = artist_ids[i0 + lane];
        sBid[lane] = album_ids[i0 + lane];
    }
    __syncthreads();
    // stage fused A tile [16 items x 64 dims] in LDS (coalesced)
    for (int idx = lane; idx < 16 * 64; idx += 32) {
        int m = idx >> 6, k = idx & 63;
        long i = i0 + m;
        sA[idx] = item_audio[i * 64 + k]
                + artist_emb[(long)sAid[m] * 64 + k]
                + album_emb[(long)sBid[m] * 64 + k];
    }
    __syncthreads();

    int ml = lane & 15;            // M for A, N for B/C
    int kg = (lane >> 4) * 2;      // K sub-offset per half-wave
    v8f c0 = {}, c1 = {}, c2 = {}, c3 = {};
    for (int k0 = 0; k0 < 64; k0 += 4) {
        v2f a;
        a.x = sA[ml * 64 + k0 + kg];
        a.y = sA[ml * 64 + k0 + kg + 1];
        // B[k,n] = W[n,k]; lane n = ml + 16*tile, contiguous K-pair from row n
        const float* w0 = Wt + (long)(ml +  0) * 64 + k0 + kg;
        const float* w1 = Wt + (long)(ml + 16) * 64 + k0 + kg;
        const float* w2 = Wt + (long)(ml + 32) * 64 + k0 + kg;
        const float* w3 = Wt + (long)(ml + 48) * 64 + k0 + kg;
        v2f b0; b0.x = w0[0]; b0.y = w0[1];
        v2f b1; b1.x = w1[0]; b1.y = w1[1];
        v2f b2; b2.x = w2[0]; b2.y = w2[1];
        v2f b3; b3.x = w3[0]; b3.y = w3[1];
        c0 = __builtin_amdgcn_wmma_f32_16x16x4_f32(false, a, false, b0, (short)0, c0, false, false);
        c1 = __builtin_amdgcn_wmma_f32_16x16x4_f32(false, a, false, b1, (short)0, c1, false, false);
        c2 = __builtin_amdgcn_wmma_f32_16x16x4_f32(false, a, false, b2, (short)0, c2, false, false);
        c3 = __builtin_amdgcn_wmma_f32_16x16x4_f32(false, a, false, b3, (short)0, c3, false, false);
    }
    __syncthreads();
    // C layout: VGPR j -> (M = j + 8*(lane>>4), N = lane&15); add bias, park in LDS
    float bi0 = bias[ml + 0], bi1 = bias[ml + 16], bi2 = bias[ml + 32], bi3 = bias[ml + 48];
#pragma unroll
    for (int j = 0; j < 8; ++j) {
        int m = j + (lane >> 4) * 8;
        sA[m * 64 + ml +  0] = c0[j] + bi0;
        sA[m * 64 + ml + 16] = c1[j] + bi1;
        sA[m * 64 + ml + 32] = c2[j] + bi2;
        sA[m * 64 + ml + 48] = c3[j] + bi3;
    }
    __syncthreads();
    // L2 norm per row: 2 lanes per row
    {
        int r = lane >> 1, half = lane & 1;
        float s = 0.0f;
        for (int k = half * 32; k < half * 32 + 32; ++k) { float v = sA[r * 64 + k]; s += v * v; }
        s += __shfl_xor(s, 1, 32);
        if (half == 0) sInv[r] = 1.0f / fmaxf(sqrtf(s), 1e-12f);
    }
    __syncthreads();
    for (int idx = lane; idx < 16 * 64; idx += 32) {
        int m = idx >> 6;
        x[(long)(NUSR + i0 + m) * 64 + (idx & 63)] = sA[idx] * sInv[m];
    }
}

// ---------------- degree scatter, rsqrt, edge normalization ----------------
__global__ void deg_kernel(const int* __restrict__ eu, const int* __restrict__ ei,
                           const float* __restrict__ w, float* __restrict__ deg) {
    int e = blockIdx.x * blockDim.x + threadIdx.x;
    if (e >= NE2) return;
    int col = (e < NE) ? (ei[e] + NUSR) : eu[e - NE];
    atomicAdd(&deg[col], w[e]);
}

__global__ void dis_kernel(float* deg) {
    int n = blockIdx.x * blockDim.x + threadIdx.x;
    if (n >= NTOT) return;
    float d = deg[n];
    deg[n] = (d > 0.0f) ? rsqrtf(d) : 0.0f;
}

__global__ void norme_kernel(const int* __restrict__ eu, const int* __restrict__ ei,
                             const float* __restrict__ dis, float* __restrict__ w) {
    int e = blockIdx.x * blockDim.x + threadIdx.x;
    if (e >= NE2) return;
    int row, col;
    if (e < NE) { row = eu[e];             col = ei[e] + NUSR; }
    else        { row = ei[e - NE] + NUSR; col = eu[e - NE];   }
    w[e] = dis[row] * w[e] * dis[col];
}

// ---------------- LGConv layer: x_dst[col] += x_src[row] * norm_e ----------------
__global__ void prop_kernel(const int* __restrict__ eu, const int* __restrict__ ei,
                            const float* __restrict__ ne,
                            const float* __restrict__ xs, float* __restrict__ xd) {
    long t = (long)blockIdx.x * blockDim.x + threadIdx.x;
    if (t >= (long)NE2 * 16) return;
    int e  = (int)(t >> 4);
    int d0 = (int)(t & 15) * 4;
    int row, col;
    if (e < NE) { row = eu[e];             col = ei[e] + NUSR; }
    else        { int k = e - NE; row = ei[k] + NUSR; col = eu[k]; }
    float s = ne[e];
    float4 v = *(const float4*)(xs + (long)row * 64 + d0);
    float* dst = xd + (long)col * 64 + d0;
    atomicAdd(dst + 0, v.x * s);
    atomicAdd(dst + 1, v.y * s);
    atomicAdd(dst + 2, v.z * s);
    atomicAdd(dst + 3, v.w * s);
}

// ---------------- row-wise L2 normalize (one wave per 64-dim row) ----------------
__global__ void l2norm_kernel(float* x, int nrows) {
    int wave = threadIdx.x >> 5;
    int lane = threadIdx.x & 31;
    int row = blockIdx.x * 8 + wave;
    if (row >= nrows) return;
    float* p = x + (long)row * 64;
    float a = p[lane], b = p[lane + 32];
    float s = a * a + b * b;
#pragma unroll
    for (int m = 16; m >= 1; m >>= 1) s += __shfl_xor(s, m, 32);
    float inv = 1.0f / fmaxf(sqrtf(s), 1e-12f);
    p[lane] = a * inv;
    p[lane + 32] = b * inv;
}

// ---------------- gather u_emb + cosine alignment loss ----------------
__global__ void gather_loss_kernel(const float* __restrict__ x,
                                   const int* __restrict__ ui, const int* __restrict__ pi,
                                   float* __restrict__ out) {
    int wave = threadIdx.x >> 5;
    int lane = threadIdx.x & 31;
    int b = blockIdx.x * 8 + wave;
    if (b >= BBAT) return;
    const float* up = x + (long)ui[b] * 64;
    const float* pp = x + (long)(NUSR + pi[b]) * 64;
    float u0 = up[lane], u1 = up[lane + 32];
    float p0 = pp[lane], p1 = pp[lane + 32];
    out[(long)b * 64 + lane] = u0;
    out[(long)b * 64 + lane + 32] = u1;
    float uu = u0 * u0 + u1 * u1;
    float pn = p0 * p0 + p1 * p1;
    float dp = u0 * p0 + u1 * p1;
#pragma unroll
    for (int m = 16; m >= 1; m >>= 1) {
        uu += __shfl_xor(uu, m, 32);
        pn += __shfl_xor(pn, m, 32);
        dp += __shfl_xor(dp, m, 32);
    }
    if (lane == 0) {
        float cosv = dp / (fmaxf(sqrtf(uu), 1e-8f) * fmaxf(sqrtf(pn), 1e-8f));
        atomicAdd(out + (long)BBAT * 64, (1.0f - cosv) * (1.0f / BBAT));
    }
}

extern "C" void kernel_launch(void* const* d_in, const int* in_sizes, int n_in,
                              void* d_out, int out_size, void* d_ws, size_t ws_size,
                              hipStream_t stream) {
    const float* user_emb   = (const float*)d_in[0];
    const float* artist_emb = (const float*)d_in[1];
    const float* album_emb  = (const float*)d_in[2];
    const float* item_audio = (const float*)d_in[3];
    const float* adapter_W  = (const float*)d_in[4];
    const float* adapter_b  = (const float*)d_in[5];
    const float* mlp_W1     = (const float*)d_in[6];
    const float* mlp_b1     = (const float*)d_in[7];
    const float* mlp_W2     = (const float*)d_in[8];
    const float* mlp_b2     = (const float*)d_in[9];
    const float* edge_feat  = (const float*)d_in[10];
    const int*   edge_user  = (const int*)d_in[11];
    const int*   edge_item  = (const int*)d_in[12];
    const int*   artist_ids = (const int*)d_in[13];
    const int*   album_ids  = (const int*)d_in[14];
    const int*   user_idx   = (const int*)d_in[15];
    const int*   pos_item   = (const int*)d_in[16];
    float* out = (float*)d_out;

    char* ws = (char*)d_ws;
    size_t xbytes = (size_t)NTOT * 64 * sizeof(float);      // 76.8 MB each
    float* xa  = (float*)(ws);
    float* xb  = (float*)(ws + xbytes);
    float* w   = (float*)(ws + 2 * xbytes);                 // 2E floats (reused as norm_e)
    float* deg = (float*)(ws + 2 * xbytes + (size_t)NE2 * sizeof(float)); // N floats (reused as dis)

    // 1) edge weights
    edge_mlp_kernel<<<(NE2 + 255) / 256, 256, 0, stream>>>(edge_feat, mlp_W1, mlp_b1, mlp_W2, mlp_b2, w);
    // 2) x0: users + (fuse -> adapter WMMA GEMM -> L2norm) items
    copy_kernel<<<2048, 256, 0, stream>>>(user_emb, xa, (long)NUSR * 64 / 4);
    fuse_adapter_kernel<<<NITM / 16, 32, 0, stream>>>(item_audio, artist_emb, album_emb,
                                                      artist_ids, album_ids, adapter_W, adapter_b, xa);
    // 3) gcn_norm: deg -> rsqrt -> norm_e (in-place over w)
    zero_kernel<<<1024, 256, 0, stream>>>(deg, NTOT);
    deg_kernel<<<(NE2 + 255) / 256, 256, 0, stream>>>(edge_user, edge_item, w, deg);
    dis_kernel<<<(NTOT + 255) / 256, 256, 0, stream>>>(deg);
    norme_kernel<<<(NE2 + 255) / 256, 256, 0, stream>>>(edge_user, edge_item, deg, w);
    // 4) three LGConv layers (ping-pong; src+dst together fit in 192MB L2)
    long pt = (long)NE2 * 16;
    int pblocks = (int)((pt + 255) / 256);
    zero_kernel<<<4096, 256, 0, stream>>>(xb, (long)NTOT * 64);
    prop_kernel<<<pblocks, 256, 0, stream>>>(edge_user, edge_item, w, xa, xb);
    zero_kernel<<<4096, 256, 0, stream>>>(xa, (long)NTOT * 64);
    prop_kernel<<<pblocks, 256, 0, stream>>>(edge_user, edge_item, w, xb, xa);
    zero_kernel<<<4096, 256, 0, stream>>>(xb, (long)NTOT * 64);
    prop_kernel<<<pblocks, 256, 0, stream>>>(edge_user, edge_item, w, xa, xb);
    // 5) final L2 normalize over all nodes
    l2norm_kernel<<<(NTOT + 7) / 8, 256, 0, stream>>>(xb, NTOT);
    // 6) outputs: u_emb [4096,64] then align_loss scalar
    zero_kernel<<<1, 32, 0, stream>>>(out + (long)BBAT * 64, 1);
    gather_loss_kernel<<<(BBAT + 7) / 8, 256, 0, stream>>>(xb, user_idx, pos_item, out);
}